// TemporalGCN_17540646437553
// MI455X (gfx1250) — compile-verified
//
#include <hip/hip_runtime.h>
#include <math.h>

// ---------------------------------------------------------------------------
// Types for WMMA fragments (f32 16x16x4: A=16x4 -> 2 VGPR, B=4x16 -> 2 VGPR,
// C/D = 16x16 -> 8 VGPR, wave32)
// ---------------------------------------------------------------------------
typedef __attribute__((ext_vector_type(2))) float v2f;
typedef __attribute__((ext_vector_type(8))) float v8f;

#define ACT_NONE 0
#define ACT_RELU 1
#define ACT_TANH 2

#define K_MAX 256
#define B_STRIDE 40   // padded LDS row stride (floats): lanes16-31 offset = 80
                      // -> +16 banks vs lanes0-15 -> conflict-free dual half-wave

// ---------------------------------------------------------------------------
// WMMA GEMM: C[M,N] = act(A[M,K] @ B[K,N] + bias[N])
// Block: 256 threads = 8 waves. Each wave computes a 16x32 output tile
// (two 16x16 WMMA accumulators sharing one A fragment). Block covers
// 128 rows x 32 cols. B panel (K x 32) staged in LDS once, coalesced.
// Requires M%16==0, N%32==0, K%4==0, K<=K_MAX.
// Fragment layouts per CDNA5 ISA 7.12.2:
//   A v0/v1: lanes 0-15 row M=lane, K=k0+{0,1}; lanes 16-31 K=k0+{2,3}
//   B v0/v1: lanes 0-15 K-rows k0+{0,1} at N=lane; lanes 16-31 k0+{2,3}
//   C  vr  : lanes 0-15 -> M=r, lanes 16-31 -> M=r+8, N = lane&15
// ---------------------------------------------------------------------------
__global__ __launch_bounds__(256) void gemm_wmma_f32_kernel(
    const float* __restrict__ A, const float* __restrict__ B,
    const float* __restrict__ bias, float* __restrict__ C,
    int M, int N, int K, int act)
{
  __shared__ float Blds[K_MAX * B_STRIDE];

  const int wave = threadIdx.x >> 5;
  const int lane = threadIdx.x & 31;
  const int half = lane >> 4;   // 0: lanes 0-15, 1: lanes 16-31
  const int l    = lane & 15;
  const int tileM = (blockIdx.x * 8 + wave) * 16;
  const int tileN = blockIdx.y * 32;

  // Cooperative coalesced load of B panel [K x 32] into LDS.
  // 32 consecutive threads fetch one contiguous 128B row chunk.
  for (int i = threadIdx.x; i < K * 32; i += 256) {
    const int k = i >> 5, c = i & 31;
    Blds[k * B_STRIDE + c] = B[(size_t)k * N + tileN + c];
  }
  __syncthreads();

  // Clamp A reads for padded tail tiles (store is skipped); keeps EXEC all-1s.
  const int tMc = (tileM < M) ? tileM : (M - 16);
  const float* __restrict__ Arow = A + (size_t)(tMc + l) * K;

  v8f acc0 = {};
  v8f acc1 = {};
  for (int k0 = 0; k0 < K; k0 += 4) {
    const int ka = k0 + 2 * half;
    v2f a;
    a.x = Arow[ka];
    a.y = Arow[ka + 1];
    v2f b0;
    b0.x = Blds[ka * B_STRIDE + l];
    b0.y = Blds[(ka + 1) * B_STRIDE + l];
    v2f b1;
    b1.x = Blds[ka * B_STRIDE + 16 + l];
    b1.y = Blds[(ka + 1) * B_STRIDE + 16 + l];
    acc0 = __builtin_amdgcn_wmma_f32_16x16x4_f32(false, a, false, b0,
                                                 (short)0, acc0, false, false);
    acc1 = __builtin_amdgcn_wmma_f32_16x16x4_f32(false, a, false, b1,
                                                 (short)0, acc1, false, false);
  }

  if (tileM >= M) return;

  const int col0 = tileN + l;
  const int col1 = tileN + 16 + l;
  const float bv0 = bias ? bias[col0] : 0.0f;
  const float bv1 = bias ? bias[col1] : 0.0f;
#pragma unroll
  for (int r = 0; r < 8; ++r) {
    const int row = tileM + r + 8 * half;
    float v0 = acc0[r] + bv0;
    float v1 = acc1[r] + bv1;
    if (act == ACT_RELU)      { v0 = fmaxf(v0, 0.0f); v1 = fmaxf(v1, 0.0f); }
    else if (act == ACT_TANH) { v0 = tanhf(v0);       v1 = tanhf(v1);       }
    C[(size_t)row * N + col0] = v0;
    C[(size_t)row * N + col1] = v1;
  }
}

// ---------------------------------------------------------------------------
// Elementwise / graph kernels
// ---------------------------------------------------------------------------
__global__ void fill_f32_kernel(float* __restrict__ p, float v, int n) {
  int i = blockIdx.x * blockDim.x + threadIdx.x;
  if (i < n) p[i] = v;
}

__global__ void deg_count_kernel(const int* __restrict__ tgt,
                                 float* __restrict__ deg, int E) {
  int e = blockIdx.x * blockDim.x + threadIdx.x;
  if (e < E) atomicAdd(&deg[tgt[e]], 1.0f);
}

__global__ void rsqrt_inplace_kernel(float* __restrict__ p, int n) {
  int i = blockIdx.x * blockDim.x + threadIdx.x;
  if (i < n) p[i] = rsqrtf(p[i]);   // deg >= 1 always (self loop)
}

// out[i,:] = xw[i,:] * dinv[i]^2   (self-loop contribution, also initializes)
__global__ void self_loop_init_kernel(const float* __restrict__ xw,
                                      const float* __restrict__ dinv,
                                      float* __restrict__ outp, int N, int H) {
  int idx = blockIdx.x * blockDim.x + threadIdx.x;
  if (idx >= N * H) return;
  int i = idx / H;
  float d = dinv[i];
  outp[idx] = xw[idx] * d * d;
}

// out[t,:] += xw[s,:] * dinv[s]*dinv[t]  for each edge (block-per-edge)
__global__ void edge_aggregate_kernel(const float* __restrict__ xw,
                                      const int* __restrict__ src,
                                      const int* __restrict__ tgt,
                                      const float* __restrict__ dinv,
                                      float* __restrict__ outp, int E, int H) {
  int e = blockIdx.x;
  if (e >= E) return;
  int s = src[e], t = tgt[e];
  float norm = dinv[s] * dinv[t];
  const float* __restrict__ xs = xw + (size_t)s * H;
  float* __restrict__ ot = outp + (size_t)t * H;
  for (int f = threadIdx.x; f < H; f += blockDim.x)
    atomicAdd(&ot[f], xs[f] * norm);
}

// out = act(in + bias[f])
__global__ void bias_act_kernel(const float* __restrict__ in,
                                const float* __restrict__ bias,
                                float* __restrict__ outp, int N, int H, int act) {
  int idx = blockIdx.x * blockDim.x + threadIdx.x;
  if (idx >= N * H) return;
  float v = in[idx] + bias[idx % H];
  if (act == ACT_RELU) v = fmaxf(v, 0.0f);
  outp[idx] = v;
}

// emb = agg + bias[f] + h1  (layer-2 output + skip connection)
__global__ void emb_write_kernel(const float* __restrict__ agg,
                                 const float* __restrict__ bias,
                                 const float* __restrict__ h1,
                                 float* __restrict__ embs, int N, int H) {
  int idx = blockIdx.x * blockDim.x + threadIdx.x;
  if (idx >= N * H) return;
  embs[idx] = agg[idx] + bias[idx % H] + h1[idx];
}

// out[row] = (sigmoid?)(dot(rows[row,:], w) + biasp[0]); one wave per row
__global__ void rowdot_kernel(const float* __restrict__ rows,
                              const float* __restrict__ w,
                              const float* __restrict__ biasp,
                              float* __restrict__ outp, int R, int Kd, int sigm) {
  int wave = threadIdx.x >> 5;
  int lane = threadIdx.x & 31;
  int row = blockIdx.x * (blockDim.x >> 5) + wave;
  if (row >= R) return;
  const float* __restrict__ a = rows + (size_t)row * Kd;
  float sum = 0.0f;
  for (int k = lane; k < Kd; k += 32) sum += a[k] * w[k];
#pragma unroll
  for (int off = 16; off > 0; off >>= 1) sum += __shfl_down(sum, off, 32);
  if (lane == 0) {
    float v = sum + biasp[0];
    if (sigm) v = 1.0f / (1.0f + expf(-v));
    outp[row] = v;
  }
}

// Per-node softmax over S=5 snapshots, fused weighted sum of embeddings.
__global__ void softmax_fuse_kernel(const float* __restrict__ emb,
                                    const float* __restrict__ scores,
                                    float* __restrict__ fused, int N, int H) {
  int n = blockIdx.x;
  if (n >= N) return;
  float sc[5];
  float m = -1e30f;
#pragma unroll
  for (int s = 0; s < 5; ++s) { sc[s] = scores[(size_t)s * N + n]; m = fmaxf(m, sc[s]); }
  float denom = 0.0f;
#pragma unroll
  for (int s = 0; s < 5; ++s) { sc[s] = expf(sc[s] - m); denom += sc[s]; }
  float inv = 1.0f / denom;
#pragma unroll
  for (int s = 0; s < 5; ++s) sc[s] *= inv;
  for (int f = threadIdx.x; f < H; f += blockDim.x) {
    float acc = 0.0f;
#pragma unroll
    for (int s = 0; s < 5; ++s)
      acc += sc[s] * emb[((size_t)s * N + n) * H + f];
    fused[(size_t)n * H + f] = acc;
  }
}

__global__ void gather_rows_kernel(const float* __restrict__ fused,
                                   const int* __restrict__ mask,
                                   float* __restrict__ post, int P, int H) {
  int idx = blockIdx.x * blockDim.x + threadIdx.x;
  if (idx >= P * H) return;
  int p = idx / H, f = idx % H;
  post[idx] = fused[(size_t)mask[p] * H + f];
}

// ---------------------------------------------------------------------------
// Launcher
// ---------------------------------------------------------------------------
static inline void launch_gemm(const float* A, const float* B, const float* bias,
                               float* C, int M, int N, int K, int act,
                               hipStream_t stream) {
  dim3 grid((M / 16 + 7) / 8, N / 32);
  gemm_wmma_f32_kernel<<<grid, 256, 0, stream>>>(A, B, bias, C, M, N, K, act);
}

extern "C" void kernel_launch(void* const* d_in, const int* in_sizes, int n_in,
                              void* d_out, int out_size, void* d_ws, size_t ws_size,
                              hipStream_t stream) {
  (void)in_sizes; (void)n_in; (void)out_size; (void)ws_size;

  const float* X    = (const float*)d_in[0];   // [N, 128]
  const int*   EI   = (const int*)  d_in[1];   // [S, 2, E]
  const int*   mask = (const int*)  d_in[2];   // [P]
  const float* W0   = (const float*)d_in[3];   // [S, 128, 256]
  const float* b0   = (const float*)d_in[4];   // [S, 256]
  const float* W1   = (const float*)d_in[5];   // [S, 256, 256]
  const float* b1   = (const float*)d_in[6];   // [S, 256]
  const float* aw1  = (const float*)d_in[7];   // [256, 128]
  const float* ab1  = (const float*)d_in[8];   // [128]
  const float* aw2  = (const float*)d_in[9];   // [128, 1]
  const float* ab2  = (const float*)d_in[10];  // [1]
  const float* cw1  = (const float*)d_in[11];  // [256, 128]
  const float* cb1  = (const float*)d_in[12];  // [128]
  const float* cw2  = (const float*)d_in[13];  // [128, 1]
  const float* cb2  = (const float*)d_in[14];  // [1]
  float* out = (float*)d_out;                  // [P]

  constexpr int N = 50000, E = 200000, S = 5, DIN = 128, H = 256, HH = 128, P = 8192;

  // Workspace layout (floats)
  float* ws = (float*)d_ws;
  size_t o = 0;
  float* emb    = ws + o; o += (size_t)S * N * H;   // 64.0M
  float* buf0   = ws + o; o += (size_t)N * H;       // xw / hw / t1 / fused
  float* buf1   = ws + o; o += (size_t)N * H;       // h1 / hcls
  float* buf2   = ws + o; o += (size_t)N * H;       // agg / post
  float* dinv   = ws + o; o += (size_t)N;
  float* scores = ws + o; o += (size_t)S * N;

  const int NH_blocks = (N * H + 255) / 256;
  const int PH_blocks = (P * H + 255) / 256;

  for (int s = 0; s < S; ++s) {
    const int* src = EI + (size_t)s * 2 * E;
    const int* tgt = src + E;

    // Degree normalization: deg = 1 (self loop) + count(tgt); dinv = rsqrt(deg)
    fill_f32_kernel<<<(N + 255) / 256, 256, 0, stream>>>(dinv, 1.0f, N);
    deg_count_kernel<<<(E + 255) / 256, 256, 0, stream>>>(tgt, dinv, E);
    rsqrt_inplace_kernel<<<(N + 255) / 256, 256, 0, stream>>>(dinv, N);

    // Layer 0: xw = X @ W0[s]  (bias added post-aggregation, per reference)
    launch_gemm(X, W0 + (size_t)s * DIN * H, nullptr, buf0, N, H, DIN, ACT_NONE, stream);
    self_loop_init_kernel<<<NH_blocks, 256, 0, stream>>>(buf0, dinv, buf2, N, H);
    edge_aggregate_kernel<<<E, 256, 0, stream>>>(buf0, src, tgt, dinv, buf2, E, H);
    bias_act_kernel<<<NH_blocks, 256, 0, stream>>>(buf2, b0 + (size_t)s * H, buf1, N, H, ACT_RELU);

    // Layer 1: hw = h1 @ W1[s]; aggregate; emb[s] = agg + b1 + h1 (skip)
    launch_gemm(buf1, W1 + (size_t)s * H * H, nullptr, buf0, N, H, H, ACT_NONE, stream);
    self_loop_init_kernel<<<NH_blocks, 256, 0, stream>>>(buf0, dinv, buf2, N, H);
    edge_aggregate_kernel<<<E, 256, 0, stream>>>(buf0, src, tgt, dinv, buf2, E, H);
    emb_write_kernel<<<NH_blocks, 256, 0, stream>>>(buf2, b1 + (size_t)s * H, buf1,
                                                    emb + (size_t)s * N * H, N, H);
  }

  // Temporal attention scores: tanh(emb @ att_w1 + att_b1) @ att_w2 + att_b2
  for (int s = 0; s < S; ++s) {
    launch_gemm(emb + (size_t)s * N * H, aw1, ab1, buf0, N, HH, H, ACT_TANH, stream);
    rowdot_kernel<<<(N + 7) / 8, 256, 0, stream>>>(buf0, aw2, ab2,
                                                   scores + (size_t)s * N, N, HH, 0);
  }

  // Softmax over snapshots + fused sum -> buf0 [N, H]
  softmax_fuse_kernel<<<N, 256, 0, stream>>>(emb, scores, buf0, N, H);

  // Classifier on gathered post rows
  gather_rows_kernel<<<PH_blocks, 256, 0, stream>>>(buf0, mask, buf2, P, H);
  launch_gemm(buf2, cw1, cb1, buf1, P, HH, H, ACT_RELU, stream);
  rowdot_kernel<<<(P + 7) / 8, 256, 0, stream>>>(buf1, cw2, cb2, out, P, HH, 1);
}